// MedianPool2d_51943334478269
// MI455X (gfx1250) — compile-verified
//
#include <hip/hip_runtime.h>
#include <hip/hip_bf16.h>
#include <stdint.h>

// Problem constants (16,3,512,512) fp32, 3x3 median, reflect pad 1.
#define IMG_H 512
#define IMG_W 512
#define TW 64               // tile width  (outputs)
#define TH 16               // tile height (outputs)
#define TILE_W (TW + 2)     // 66 (halo)
#define TILE_H (TH + 2)     // 18 (halo)
#define NTHREADS 256        // 8 waves of 32

__device__ __forceinline__ void s2(float& a, float& b) {
    float t = fminf(a, b);
    b = fmaxf(a, b);
    a = t;
}

// McGuire median-of-9 exchange network (mnmx6 -> mnmx5 -> mnmx4 -> mnmx3).
// Dead extremes are DCE'd by the compiler; ~24 v_min/v_max ops survive.
__device__ __forceinline__ float median9(float v0, float v1, float v2,
                                         float v3, float v4, float v5,
                                         float v6, float v7, float v8) {
    // mnmx6(v0..v5): v0=min of 6, v5=max of 6 -> discarded
    s2(v0, v3); s2(v1, v4); s2(v2, v5);
    s2(v0, v1); s2(v0, v2);
    s2(v4, v5); s2(v3, v5);
    // mnmx5(v1,v2,v3,v4,v6): v1=min, v6=max -> discarded
    s2(v1, v2); s2(v3, v4);
    s2(v1, v3); s2(v1, v6);
    s2(v4, v6); s2(v2, v6);
    // mnmx4(v2,v3,v4,v7): v2=min, v7=max -> discarded
    s2(v2, v3); s2(v4, v7); s2(v2, v4); s2(v3, v7);
    // mnmx3(v3,v4,v8): v3=min, v8(max) -> discarded; v4 = median
    s2(v4, v8); s2(v3, v8); s2(v3, v4);
    return v4;
}

__global__ __launch_bounds__(NTHREADS)
void MedianPool2d_51943334478269_kernel(const float* __restrict__ x,
                                        float* __restrict__ out) {
    __shared__ float tile[TILE_H][TILE_W];

    const int tid = threadIdx.x;
    const int x0  = blockIdx.x * TW;
    const int y0  = blockIdx.y * TH;
    const int p   = blockIdx.z;                  // plane index (B*C)

    const float* plane = x + (size_t)p * (IMG_H * IMG_W);
    const uint64_t plane_addr = (uint64_t)(uintptr_t)plane;

    // ---- Stage halo tile (66x18) into LDS via CDNA5 async global->LDS DMA.
    // Reflect padding resolved per lane during the gather.
    for (int idx = tid; idx < TILE_H * TILE_W; idx += NTHREADS) {
        int r  = idx / TILE_W;
        int c  = idx - r * TILE_W;
        int gy = y0 - 1 + r;
        int gx = x0 - 1 + c;
        gy = (gy < 0) ? -gy : ((gy >= IMG_H) ? (2 * IMG_H - 2 - gy) : gy);
        gx = (gx < 0) ? -gx : ((gx >= IMG_W) ? (2 * IMG_W - 2 - gx) : gx);
        uint64_t gaddr = plane_addr + ((uint64_t)(gy * IMG_W + gx) << 2);
        uint32_t laddr = (uint32_t)(uintptr_t)&tile[r][c];  // LDS byte offset
        asm volatile("global_load_async_to_lds_b32 %0, %1, off"
                     :: "v"(laddr), "v"(gaddr)
                     : "memory");
    }
    // Our inline-asm async ops are tracked by ASYNCcnt, which the compiler's
    // automatic waitcnt insertion does not know about: wait explicitly, then
    // barrier so every wave sees the fully populated tile.
    asm volatile("s_wait_asynccnt 0" ::: "memory");
    __syncthreads();

    // ---- Compute: each thread handles 4 consecutive rows of one column,
    // keeping a rolling 3x3 window in registers (18 LDS loads / 4 outputs).
    const int lx = tid & (TW - 1);        // 0..63
    const int ry = (tid >> 6) << 2;       // 0,4,8,12 (tile-row of first output)

    float* oplane = out + (size_t)p * (IMG_H * IMG_W);

    float a0 = tile[ry + 0][lx], a1 = tile[ry + 0][lx + 1], a2 = tile[ry + 0][lx + 2];
    float b0 = tile[ry + 1][lx], b1 = tile[ry + 1][lx + 1], b2 = tile[ry + 1][lx + 2];
    float c0 = tile[ry + 2][lx], c1 = tile[ry + 2][lx + 1], c2 = tile[ry + 2][lx + 2];

#pragma unroll
    for (int i = 0; i < 4; ++i) {
        float m = median9(a0, a1, a2, b0, b1, b2, c0, c1, c2);
        // Streaming output, never re-read: non-temporal store.
        __builtin_nontemporal_store(m, &oplane[(size_t)(y0 + ry + i) * IMG_W + x0 + lx]);
        if (i < 3) {
            a0 = b0; a1 = b1; a2 = b2;
            b0 = c0; b1 = c1; b2 = c2;
            c0 = tile[ry + 3 + i][lx];
            c1 = tile[ry + 3 + i][lx + 1];
            c2 = tile[ry + 3 + i][lx + 2];
        }
    }
}

extern "C" void kernel_launch(void* const* d_in, const int* in_sizes, int n_in,
                              void* d_out, int out_size, void* d_ws, size_t ws_size,
                              hipStream_t stream) {
    (void)n_in; (void)d_ws; (void)ws_size; (void)out_size;
    const float* x = (const float*)d_in[0];
    float* out     = (float*)d_out;

    const int planes = in_sizes[0] / (IMG_H * IMG_W);   // 16*3 = 48
    dim3 grid(IMG_W / TW, IMG_H / TH, planes);          // (8, 32, 48)
    dim3 block(NTHREADS);
    MedianPool2d_51943334478269_kernel<<<grid, block, 0, stream>>>(x, out);
}